// ZOHLinearLayer_38336878084731
// MI455X (gfx1250) — compile-verified
//
#include <hip/hip_runtime.h>

// ZOH SSM kernel K_bar on MI455X (gfx1250, wave32, WMMA).
//
// Pipeline (all fp32, all GEMMs on V_WMMA_F32_16X16X4_F32):
//   X = delta*A
//   A_bar = I + X(I + X/2(I + X/3(I + X/4)))              [3 GEMMs]
//   M = A + stab*I ; Minv via Newton-Schulz, 36 iters     [72 GEMMs]
//   B_bar = Minv * (delta*(A_bar*B - B))                  [2 matvecs]
//   V[:,0:128] = A_bar^i B_bar  via doubling              [7+7 GEMMs]
//   U[0:64,:]  = C * (A_bar^128)^j via doubling           [6+5 GEMMs]
//   K = U*V (64x128), emit reversed                       [1 GEMM]
//
// GEMM: 64x128x32 workgroup tile, 4 waves, 8 WMMA accumulators per wave.
// Both LDS operand tiles are stored K-innermost (+4 pad) so every WMMA
// fragment is one aligned ds_load_b64 (no v_mov shuffles, batched waits).
// Next K-tile is prefetched with global_prefetch_b8 while computing.

typedef __attribute__((ext_vector_type(2))) float v2f;
typedef __attribute__((ext_vector_type(8))) float v8f;

#define BM 64
#define BN 128
#define BK 32
#define KPAD (BK + 4)      // 16-B aligned rows, conflict-free frag reads
#define GEMM_THREADS 128   // 4 waves of 32

// C[M,N] = alpha * A[M,K] * B[K,N] (+ I if addI). Row-major, leading dims.
__global__ __launch_bounds__(GEMM_THREADS)
void wmma_gemm_f32(const float* __restrict__ A, const float* __restrict__ B,
                   float* __restrict__ C, int Mdim, int Ndim, int Kdim,
                   int lda, int ldb, int ldc, float alpha, int addI) {
  __shared__ float As[BM][KPAD];   // A tile, K-innermost
  __shared__ float Bt[BN][KPAD];   // B tile, transposed to K-innermost

  const int tid  = threadIdx.x;
  const int wave = tid >> 5;
  const int lane = tid & 31;
  const int half = lane >> 4;     // 0: K pair {0,1}, rows 0-7 ; 1: {2,3}, rows 8-15
  const int l16  = lane & 15;
  const int rowBase = blockIdx.y * BM;
  const int colBase = blockIdx.x * BN;
  const int aRow = wave * 16 + l16;

  v8f acc[8] = {v8f{}, v8f{}, v8f{}, v8f{}, v8f{}, v8f{}, v8f{}, v8f{}};

  for (int k0 = 0; k0 < Kdim; k0 += BK) {
    // ---- prefetch next K-tile into cache (global_prefetch_b8) ----
    if (k0 + BK < Kdim) {
      if (tid < BM) {
        int gr = rowBase + tid;
        if (gr < Mdim)
          __builtin_prefetch(&A[(size_t)gr * lda + k0 + BK], 0, 3);
      }
      int br = k0 + BK + (tid >> 2);
      int bc = colBase + (tid & 3) * 32;
      if (br < Kdim && bc < Ndim)
        __builtin_prefetch(&B[(size_t)br * ldb + bc], 0, 3);
    }

    // ---- stage A tile 64x32: float4 global -> float4 LDS ----
    #pragma unroll
    for (int j = 0; j < 4; ++j) {
      int idx = tid + j * GEMM_THREADS;          // 0..511
      int r = idx >> 3, cq = idx & 7;            // row, float4-column
      int gr = rowBase + r, gc = k0 + cq * 4;
      float4 v = make_float4(0.f, 0.f, 0.f, 0.f);
      if (gr < Mdim) {
        const float* p = &A[(size_t)gr * lda + gc];
        if (gc + 3 < Kdim) v = *(const float4*)p;
        else {
          if (gc + 0 < Kdim) v.x = p[0];
          if (gc + 1 < Kdim) v.y = p[1];
          if (gc + 2 < Kdim) v.z = p[2];
          if (gc + 3 < Kdim) v.w = p[3];
        }
      }
      *(float4*)&As[r][cq * 4] = v;
    }

    // ---- stage B tile 32x128: float4 global -> transposed LDS ----
    #pragma unroll
    for (int j = 0; j < 8; ++j) {
      int idx = tid + j * GEMM_THREADS;          // 0..1023
      int r = idx >> 5, cq = idx & 31;           // B-row (K), float4-column
      int gr = k0 + r, gc = colBase + cq * 4;
      float4 v = make_float4(0.f, 0.f, 0.f, 0.f);
      if (gr < Kdim) {
        const float* p = &B[(size_t)gr * ldb + gc];
        if (gc + 3 < Ndim) v = *(const float4*)p;
        else {
          if (gc + 0 < Ndim) v.x = p[0];
          if (gc + 1 < Ndim) v.y = p[1];
          if (gc + 2 < Ndim) v.z = p[2];
          if (gc + 3 < Ndim) v.w = p[3];
        }
      }
      int c = cq * 4;
      Bt[c + 0][r] = v.x;
      Bt[c + 1][r] = v.y;
      Bt[c + 2][r] = v.z;
      Bt[c + 3][r] = v.w;
    }
    __syncthreads();

    // ---- compute: 8 kk-steps x (1 A-frag + 8 B-frags + 8 WMMA) ----
    #pragma unroll
    for (int kk = 0; kk < BK; kk += 4) {
      const int kb = kk + 2 * half;
      v2f a = *(const v2f*)&As[aRow][kb];        // single ds_load_b64
      #pragma unroll
      for (int t = 0; t < 8; ++t) {
        v2f b = *(const v2f*)&Bt[t * 16 + l16][kb];  // single ds_load_b64
        acc[t] = __builtin_amdgcn_wmma_f32_16x16x4_f32(
            /*neg_a=*/false, a, /*neg_b=*/false, b,
            /*c_mod=*/(short)0, acc[t], /*reuse_a=*/false, /*reuse_b=*/false);
      }
    }
    __syncthreads();
  }

  // C/D layout: vgpr i holds row i (lanes 0-15) / row i+8 (lanes 16-31), col l16.
  #pragma unroll
  for (int t = 0; t < 8; ++t) {
    #pragma unroll
    for (int i = 0; i < 8; ++i) {
      int r = rowBase + wave * 16 + half * 8 + i;
      int c = colBase + t * 16 + l16;
      if (r < Mdim && c < Ndim) {
        float v = alpha * acc[t][i];
        if (addI && r == c) v += 1.0f;
        C[(size_t)r * ldc + c] = v;
      }
    }
  }
}

// y = Mmat (n x n, row-major) * x ; one wave32 per row, shfl reduction.
__global__ void matvec_kernel(const float* __restrict__ Mmat,
                              const float* __restrict__ x,
                              float* __restrict__ y, int n) {
  int row  = blockIdx.x * (blockDim.x >> 5) + (threadIdx.x >> 5);
  int lane = threadIdx.x & 31;
  if (row >= n) return;                       // whole-wave uniform
  float s = 0.0f;
  for (int j = lane; j < n; j += 32) s += Mmat[(size_t)row * n + j] * x[j];
  #pragma unroll
  for (int off = 16; off > 0; off >>= 1) s += __shfl_down(s, off, 32);
  if (lane == 0) y[row] = s;
}

// dst = alpha*src + diag*I  (n x n elementwise)
__global__ void affine_diag_kernel(float* __restrict__ dst,
                                   const float* __restrict__ src,
                                   float alpha, float diag, int n) {
  size_t idx = (size_t)blockIdx.x * blockDim.x + threadIdx.x;
  size_t total = (size_t)n * n;
  if (idx >= total) return;
  int i = (int)(idx / (size_t)n);
  int j = (int)(idx % (size_t)n);
  float v = alpha * src[idx];
  if (i == j) v += diag;
  dst[idx] = v;
}

__global__ void zero_kernel(float* p) { *p = 0.0f; }

// out += sum(src[i]^2), grid-stride + wave reduce + atomic.
__global__ void sqsum_kernel(const float* __restrict__ src,
                             float* __restrict__ out, size_t total) {
  size_t idx    = (size_t)blockIdx.x * blockDim.x + threadIdx.x;
  size_t stride = (size_t)gridDim.x * blockDim.x;
  float s = 0.0f;
  for (size_t i = idx; i < total; i += stride) { float v = src[i]; s += v * v; }
  #pragma unroll
  for (int off = 16; off > 0; off >>= 1) s += __shfl_down(s, off, 32);
  if ((threadIdx.x & 31) == 0) atomicAdd(out, s);
}

// dst = src^T / (*sumsq)  (Newton-Schulz seed X0 = M^T/||M||_F^2)
__global__ void transpose_rscale_kernel(const float* __restrict__ src,
                                        const float* __restrict__ sumsq,
                                        float* __restrict__ dst, int n) {
  size_t idx = (size_t)blockIdx.x * blockDim.x + threadIdx.x;
  size_t total = (size_t)n * n;
  if (idx >= total) return;
  int i = (int)(idx / (size_t)n);
  int j = (int)(idx % (size_t)n);
  dst[idx] = src[(size_t)j * n + i] * (1.0f / (*sumsq));
}

// r = scale*(a - b)   (vectors, length n)
__global__ void residual_scale_kernel(float* __restrict__ r,
                                      const float* __restrict__ a,
                                      const float* __restrict__ b,
                                      float scale, int n) {
  int i = blockIdx.x * blockDim.x + threadIdx.x;
  if (i < n) r[i] = scale * (a[i] - b[i]);
}

// dst[i*stride] = src[i]
__global__ void copy_strided_kernel(float* __restrict__ dst,
                                    const float* __restrict__ src,
                                    int n, int stride) {
  int i = blockIdx.x * blockDim.x + threadIdx.x;
  if (i < n) dst[(size_t)i * stride] = src[i];
}

// out[total-1-t] = Kt[t]  (time reversal of K_bar)
__global__ void emit_kernel(const float* __restrict__ Kt,
                            float* __restrict__ out, int total) {
  int t = blockIdx.x * blockDim.x + threadIdx.x;
  if (t < total) out[total - 1 - t] = Kt[t];
}

extern "C" void kernel_launch(void* const* d_in, const int* in_sizes, int n_in,
                              void* d_out, int out_size, void* d_ws, size_t ws_size,
                              hipStream_t stream) {
  const float* A  = (const float*)d_in[0];   // n x n
  const float* Bv = (const float*)d_in[1];   // n x 1
  const float* Cv = (const float*)d_in[2];   // 1 x n
  float* out = (float*)d_out;

  const int   n     = in_sizes[1];           // 2048
  const int   S     = 128;                   // V-block length
  const int   J     = out_size / S;          // 64 blocks; out_size = 8192
  const float DELTA = 0.01f;
  const float STAB  = 1e-3f;
  const int   NS_ITERS = 36;                 // even => ping ends where it started

  float* ws = (float*)d_ws;
  const size_t nn = (size_t)n * n;
  float* W0 = ws;               // X / scratch T / P-pingpong
  float* W1 = ws + 1 * nn;      // Horner T / NS pong
  float* W2 = ws + 2 * nn;      // A_bar / P-pingpong
  float* W3 = ws + 3 * nn;      // M = A + stab*I
  float* W4 = ws + 4 * nn;      // NS ping (X_k)
  float* V    = ws + 5 * nn;                    // n x S, row-major (ld = S)
  float* U    = V + (size_t)n * S;              // J x n, row-major (ld = n)
  float* Kt   = U + (size_t)J * n;              // J x S
  float* v1   = Kt + (size_t)J * S;             // n
  float* r0   = v1 + n;                         // n
  float* Bbar = r0 + n;                         // n
  float* ssq  = Bbar + n;                       // 1

  const int EW_THREADS = 256;
  const int ewBlocks   = (int)((nn + EW_THREADS - 1) / EW_THREADS);
  const int vecBlocks  = (n + EW_THREADS - 1) / EW_THREADS;
  const int mvBlocks   = (n + 7) / 8;           // 8 waves/block, one row/wave

  auto gemm = [&](float* Cm, const float* Am, const float* Bm,
                  int M_, int N_, int K_, int lda, int ldb, int ldc,
                  float alpha, int addI) {
    dim3 grid((N_ + BN - 1) / BN, (M_ + BM - 1) / BM);
    wmma_gemm_f32<<<grid, dim3(GEMM_THREADS), 0, stream>>>(
        Am, Bm, Cm, M_, N_, K_, lda, ldb, ldc, alpha, addI);
  };

  // ---- 1. X = delta*A ; A_bar = degree-4 Taylor via Horner -------------
  affine_diag_kernel<<<ewBlocks, EW_THREADS, 0, stream>>>(W0, A, DELTA, 0.0f, n);
  affine_diag_kernel<<<ewBlocks, EW_THREADS, 0, stream>>>(W1, W0, 0.25f, 1.0f, n); // T1=I+X/4
  gemm(W2, W0, W1, n, n, n, n, n, n, 1.0f / 3.0f, 1);   // T2 = I + X*T1/3
  gemm(W1, W0, W2, n, n, n, n, n, n, 0.5f,        1);   // T3 = I + X*T2/2
  gemm(W2, W0, W1, n, n, n, n, n, n, 1.0f,        1);   // A_bar = I + X*T3

  // ---- 2. M = A + stab*I ; Minv via Newton-Schulz ----------------------
  affine_diag_kernel<<<ewBlocks, EW_THREADS, 0, stream>>>(W3, A, 1.0f, STAB, n);
  zero_kernel<<<1, 1, 0, stream>>>(ssq);
  sqsum_kernel<<<1024, EW_THREADS, 0, stream>>>(W3, ssq, nn);
  transpose_rscale_kernel<<<ewBlocks, EW_THREADS, 0, stream>>>(W3, ssq, W4, n); // X0

  float* ping = W4;
  float* pong = W1;
  for (int it = 0; it < NS_ITERS; ++it) {
    gemm(W0, W3, ping, n, n, n, n, n, n, 1.0f, 0);                        // T = M*Xk
    affine_diag_kernel<<<ewBlocks, EW_THREADS, 0, stream>>>(W0, W0, -1.0f, 2.0f, n); // 2I-T
    gemm(pong, ping, W0, n, n, n, n, n, n, 1.0f, 0);                      // Xk+1
    float* tmp = ping; ping = pong; pong = tmp;
  }
  // Minv = ping

  // ---- 3. B_bar = Minv * (delta*(A_bar*B - B)) -------------------------
  matvec_kernel<<<mvBlocks, EW_THREADS, 0, stream>>>(W2, Bv, v1, n);
  residual_scale_kernel<<<vecBlocks, EW_THREADS, 0, stream>>>(r0, v1, Bv, DELTA, n);
  matvec_kernel<<<mvBlocks, EW_THREADS, 0, stream>>>(ping, r0, Bbar, n);

  // ---- 4. V[:,i] = A_bar^i B_bar, i<128, via doubling; P -> A_bar^128 --
  copy_strided_kernel<<<vecBlocks, EW_THREADS, 0, stream>>>(V, Bbar, n, S);
  float* pcur = W2;   // A_bar
  float* palt = W0;
  for (int j = 0; j < 7; ++j) {
    int k = 1 << j;
    gemm(V + k, pcur, V, n, k, n, n, S, S, 1.0f, 0);   // V[:,k:2k) = P_k * V[:,0:k)
    gemm(palt, pcur, pcur, n, n, n, n, n, n, 1.0f, 0); // P_{2k} = P_k^2
    float* tmp = pcur; pcur = palt; palt = tmp;
  }
  // pcur = A_bar^128

  // ---- 5. U[j,:] = C * P^j, j<64, via doubling -------------------------
  copy_strided_kernel<<<vecBlocks, EW_THREADS, 0, stream>>>(U, Cv, n, 1);
  for (int j = 0; j < 6; ++j) {
    int k = 1 << j;
    gemm(U + (size_t)k * n, U, pcur, k, n, n, n, n, n, 1.0f, 0); // U[k:2k) = U[0:k)*Q_k
    if (j < 5) {
      gemm(palt, pcur, pcur, n, n, n, n, n, n, 1.0f, 0);         // Q_{2k} = Q_k^2
      float* tmp = pcur; pcur = palt; palt = tmp;
    }
  }

  // ---- 6. K = U*V (64x128), emit time-reversed -------------------------
  gemm(Kt, U, V, J, S, n, n, S, S, 1.0f, 0);
  emit_kernel<<<(out_size + EW_THREADS - 1) / EW_THREADS, EW_THREADS, 0, stream>>>(
      Kt, out, out_size);
}